// Fp8Padding_1417339207908
// MI455X (gfx1250) — compile-verified
//
#include <hip/hip_runtime.h>

// ---------------------------------------------------------------------------
// Fp8Padding row-gather + alignment-pad copy for MI455X (gfx1250).
//
// Static config (from the reference):
//   M_SPLITS        = {4091, 8177, 2045, 4093, 6133, 1021, 4085, 8189}
//   PADDED_M_SPLITS = {4096, 8192, 2048, 4096, 6144, 1024, 4096, 8192}
//   HIDDEN = 2048 (fp32)  TOTAL_IN = 37834 rows  TOTAL_OUT = 37888 rows
//
// Pure streaming copy: ~620 MB of HBM traffic, 0 FLOPs -> bandwidth bound.
// Floor at 23.3 TB/s is ~27 us. Strategy: b128 loads/stores, non-temporal
// temporal hints (stream >> 192 MB L2), one contiguous 8 KB row per block.
// ---------------------------------------------------------------------------

#define HIDDEN      2048
#define VEC4_ROW    (HIDDEN / 4)   // 512 float4 per row
#define TOTAL_OUT   37888
#define BLOCK       256            // 8 wave32s; 2 float4 per thread per row

typedef float __attribute__((ext_vector_type(4))) v4f;

__global__ __launch_bounds__(BLOCK) void fp8_pad_rows_kernel(
    const v4f* __restrict__ in, v4f* __restrict__ out)
{
    const int row = blockIdx.x;          // uniform -> SALU compare chain below

    // Prefix sums of PADDED_M_SPLITS (dst) and M_SPLITS (src), compile-time.
    int dstBase, srcBase, mrows;
    if      (row <  4096) { dstBase =     0; srcBase =     0; mrows = 4091; }
    else if (row < 12288) { dstBase =  4096; srcBase =  4091; mrows = 8177; }
    else if (row < 14336) { dstBase = 12288; srcBase = 12268; mrows = 2045; }
    else if (row < 18432) { dstBase = 14336; srcBase = 14313; mrows = 4093; }
    else if (row < 24576) { dstBase = 18432; srcBase = 18406; mrows = 6133; }
    else if (row < 25600) { dstBase = 24576; srcBase = 24539; mrows = 1021; }
    else if (row < 29696) { dstBase = 25600; srcBase = 25560; mrows = 4085; }
    else                  { dstBase = 29696; srcBase = 29645; mrows = 8189; }

    const int r = row - dstBase;                       // row within segment
    const int t = threadIdx.x;
    v4f* __restrict__ dst = out + (size_t)row * VEC4_ROW;

    if (r < mrows) {
        // Valid data row: stream-copy 8 KB. Two b128 loads in flight before
        // the stores (MLP), all non-temporal (read-once / write-once).
        const v4f* __restrict__ src = in + (size_t)(srcBase + r) * VEC4_ROW;
        v4f a = __builtin_nontemporal_load(src + t);
        v4f b = __builtin_nontemporal_load(src + t + BLOCK);
        __builtin_nontemporal_store(a, dst + t);
        __builtin_nontemporal_store(b, dst + t + BLOCK);
    } else {
        // Alignment pad row (54 total): zero-fill.
        v4f z = (v4f)0.0f;
        __builtin_nontemporal_store(z, dst + t);
        __builtin_nontemporal_store(z, dst + t + BLOCK);
    }
}

extern "C" void kernel_launch(void* const* d_in, const int* in_sizes, int n_in,
                              void* d_out, int out_size, void* d_ws, size_t ws_size,
                              hipStream_t stream)
{
    (void)in_sizes; (void)n_in; (void)out_size; (void)d_ws; (void)ws_size;
    const v4f* in  = (const v4f*)d_in[0];   // [37834, 2048] fp32
    v4f*       out = (v4f*)d_out;           // [37888, 2048] fp32

    fp8_pad_rows_kernel<<<TOTAL_OUT, BLOCK, 0, stream>>>(in, out);
}